// SparseRepresentationLoss_4466765988331
// MI455X (gfx1250) — compile-verified
//
#include <hip/hip_runtime.h>

// SparseRepresentationLoss (OMP pursuit) for MI455X / gfx1250.
//
// Strategy: Gram-matrix OMP.
//   G  = dict @ dict^T        [K,K]  (f32 WMMA 16x16x4)   -- 64 MB, lives in L2 (192 MB)
//   IP = x @ dict^T           [B,K]  (f32 WMMA 16x16x4)   -- 128 MB
//   per-sample pursuit: ip_{t+1} = ip_t - ip_t[idx]*G[idx,:], ip in LDS (16 KB)
// Workspace requirement: 192 MB (G + IP).

typedef float v2f __attribute__((ext_vector_type(2)));
typedef float v8f __attribute__((ext_vector_type(8)));

#define B_SZ  8192
#define D_DIM 1024
#define K_DIM 4096
#define SP    16

// ---------------------------------------------------------------------------
// C[n, m] = sum_d A[m,d] * B[n,d]   (C row-major [N, ldc])
// One wave per 32x32 output tile: 4 accumulators (2x2 of 16x16), so each pair
// of fragment loads (2 A-frags + 2 B-frags) feeds 4 WMMAs -> half the load
// issue and half the L2 traffic per output vs a 16x16 tile.
// A fragment (16x4, MxK): lane l<16 holds A[m0+l, d+0..1], lane l+16 holds A[m0+l, d+2..3].
// B fragment (4x16, KxN): mirrored with N in place of M.
// C/D fragment: VGPR r, lanes 0-15 -> (M=r, N=lane); lanes 16-31 -> (M=r+8, N=lane-16)
//   => per lane, c[0..7] are 8 consecutive M values -> one contiguous 32B store.
// ---------------------------------------------------------------------------
__global__ __launch_bounds__(32) void gemm_nt_wmma_f32(
    const float* __restrict__ A, const float* __restrict__ Bm,
    float* __restrict__ C, int Dk, int ldc)
{
    const int m0   = blockIdx.x * 32;
    const int n0   = blockIdx.y * 32;
    const int lane = threadIdx.x & 31;
    const int half = lane >> 4;   // which K-pair this lane supplies
    const int l    = lane & 15;   // M (for A) / N (for B)

    const float* arow0 = A  + (size_t)(m0 + l) * Dk + 2 * half;
    const float* arow1 = arow0 + (size_t)16 * Dk;
    const float* brow0 = Bm + (size_t)(n0 + l) * Dk + 2 * half;
    const float* brow1 = brow0 + (size_t)16 * Dk;

    v8f c00 = {}, c01 = {}, c10 = {}, c11 = {};
#pragma unroll 4
    for (int d = 0; d < Dk; d += 4) {
        v2f a0 = *(const v2f*)(arow0 + d);
        v2f a1 = *(const v2f*)(arow1 + d);
        v2f b0 = *(const v2f*)(brow0 + d);
        v2f b1 = *(const v2f*)(brow1 + d);
        c00 = __builtin_amdgcn_wmma_f32_16x16x4_f32(false, a0, false, b0, (short)0, c00, false, false);
        c10 = __builtin_amdgcn_wmma_f32_16x16x4_f32(false, a1, false, b0, (short)0, c10, false, false);
        c01 = __builtin_amdgcn_wmma_f32_16x16x4_f32(false, a0, false, b1, (short)0, c01, false, false);
        c11 = __builtin_amdgcn_wmma_f32_16x16x4_f32(false, a1, false, b1, (short)0, c11, false, false);
    }

    float* out0 = C + (size_t)(n0 + l) * ldc + (m0 + 8 * half);
    float* out1 = C + (size_t)(n0 + 16 + l) * ldc + (m0 + 8 * half);
    *(v8f*)(out0)      = c00;   // 32B-aligned: m0%32==0, ldc%8==0, base 256B-aligned
    *(v8f*)(out0 + 16) = c10;
    *(v8f*)(out1)      = c01;
    *(v8f*)(out1 + 16) = c11;
}

// ---------------------------------------------------------------------------
// Per-sample greedy pursuit. One 256-thread block (8 waves) per sample.
// LDS: ip[K] (16 KB) + reductions. All hot loops use 16B (b128) accesses.
// Applies updates for t=0..14 only, so after the loop LDS ip == ip_15 == last_ip
// (exactly what the reference's scan carries out of the final step).
// ---------------------------------------------------------------------------
__global__ __launch_bounds__(256) void omp_pursuit(
    const float* __restrict__ x, const float* __restrict__ dict,
    const float* __restrict__ G, const float* __restrict__ IP,
    float* __restrict__ out)
{
    __shared__ __align__(16) float ip[K_DIM];
    __shared__ float redv[256];
    __shared__ int   redi[256];
    __shared__ int   idxs[SP];
    __shared__ float cvals[SP];   // c_t = ip_t[idx_t] (residual coefficients)
    __shared__ float wvals[SP];   // diff weights
    __shared__ float sabs_s;

    const int b   = blockIdx.x;
    const int tid = threadIdx.x;

    // load ip column (float4: 4096 floats = 1024 float4s, 4 per thread)
    {
        const float4* src = (const float4*)(IP + (size_t)b * K_DIM);
        float4*       dst = (float4*)ip;
        for (int q = tid; q < K_DIM / 4; q += 256) dst[q] = src[q];
    }
    __syncthreads();

    for (int t = 0; t < SP; ++t) {
        // argmax |ip|, first-index tie-break (matches jnp.argmax):
        // within-thread scan ascends in k with strict '>', cross-thread
        // reduction breaks ties by lower index.
        float best = -1.0f; int bi = 0;
        const float4* ipv = (const float4*)ip;
        for (int q = tid; q < K_DIM / 4; q += 256) {
            float4 v4 = ipv[q];
            float a0 = fabsf(v4.x), a1 = fabsf(v4.y), a2 = fabsf(v4.z), a3 = fabsf(v4.w);
            if (a0 > best) { best = a0; bi = 4 * q + 0; }
            if (a1 > best) { best = a1; bi = 4 * q + 1; }
            if (a2 > best) { best = a2; bi = 4 * q + 2; }
            if (a3 > best) { best = a3; bi = 4 * q + 3; }
        }
        redv[tid] = best; redi[tid] = bi;
        __syncthreads();
        for (int s = 128; s > 0; s >>= 1) {
            if (tid < s) {
                float v2 = redv[tid + s]; int i2 = redi[tid + s];
                if (v2 > redv[tid] || (v2 == redv[tid] && i2 < redi[tid])) {
                    redv[tid] = v2; redi[tid] = i2;
                }
            }
            __syncthreads();
        }
        if (tid == 0) { idxs[t] = redi[0]; cvals[t] = ip[redi[0]]; }
        __syncthreads();

        if (t < SP - 1) {
            const int   idx = idxs[t];
            const float c   = cvals[t];
            const float4* grow = (const float4*)(G + (size_t)idx * K_DIM); // 16 KB, L2-resident
            float4* ipw = (float4*)ip;
            for (int q = tid; q < K_DIM / 4; q += 256) {
                float4 g = grow[q];
                float4 v = ipw[q];
                v.x -= c * g.x; v.y -= c * g.y; v.z -= c * g.z; v.w -= c * g.w;
                ipw[q] = v;
            }
            __syncthreads();
        }
    }

    // coeff values come from ip_15 at the selected indices; duplicates count once.
    if (tid == 0) {
        float sabs = 0.0f;
        for (int t = 0; t < SP; ++t) {
            int idx = idxs[t];
            bool first = true;
            for (int t2 = 0; t2 < t; ++t2)
                if (idxs[t2] == idx) { first = false; break; }
            float b15 = ip[idx];
            // diff = recon - res = sum_t (first?ip15[idx_t]:0 + c_t) * dict[idx_t] - x
            wvals[t] = cvals[t] + (first ? b15 : 0.0f);
            if (first) sabs += fabsf(b15);
        }
        sabs_s = sabs;
    }
    __syncthreads();

    // SSE of diff over D (float4: 1024 floats = 256 float4s, 1 per thread)
    float sse = 0.0f;
    {
        const float4* xv = (const float4*)(x + (size_t)b * D_DIM);
        for (int q = tid; q < D_DIM / 4; q += 256) {
            float4 xq = xv[q];
            float d0 = -xq.x, d1 = -xq.y, d2 = -xq.z, d3 = -xq.w;
#pragma unroll
            for (int t = 0; t < SP; ++t) {
                const float4 dq = ((const float4*)(dict + (size_t)idxs[t] * D_DIM))[q];
                const float w = wvals[t];
                d0 += w * dq.x; d1 += w * dq.y; d2 += w * dq.z; d3 += w * dq.w;
            }
            sse += d0 * d0 + d1 * d1 + d2 * d2 + d3 * d3;
        }
    }
    redv[tid] = sse;
    __syncthreads();
    for (int s = 128; s > 0; s >>= 1) {
        if (tid < s) redv[tid] += redv[tid + s];
        __syncthreads();
    }
    if (tid == 0) {
        float contrib = redv[0] / ((float)B_SZ * (float)D_DIM)
                      + 0.1f * sabs_s / (float)B_SZ;
        atomicAdd(out, contrib);
    }
}

__global__ void zero_out_kernel(float* out) { out[0] = 0.0f; }

extern "C" void kernel_launch(void* const* d_in, const int* in_sizes, int n_in,
                              void* d_out, int out_size, void* d_ws, size_t ws_size,
                              hipStream_t stream) {
    const float* x    = (const float*)d_in[0];   // [8192, 1024] f32
    const float* dict = (const float*)d_in[1];   // [4096, 1024] f32
    // d_in[2] = sparsity (==16, baked in as SP)
    float* out = (float*)d_out;                  // scalar loss

    float* G  = (float*)d_ws;                                        // 64 MB
    float* IP = (float*)((char*)d_ws + (size_t)K_DIM * K_DIM * 4);   // 128 MB

    zero_out_kernel<<<1, 1, 0, stream>>>(out);

    // G = dict * dict^T   [K, K]
    gemm_nt_wmma_f32<<<dim3(K_DIM / 32, K_DIM / 32), 32, 0, stream>>>(
        dict, dict, G, D_DIM, K_DIM);

    // IP = x * dict^T stored [B, K] (sample-major for the pursuit kernel)
    gemm_nt_wmma_f32<<<dim3(K_DIM / 32, B_SZ / 32), 32, 0, stream>>>(
        dict, x, IP, D_DIM, K_DIM);

    omp_pursuit<<<B_SZ, 256, 0, stream>>>(x, dict, G, IP, out);
}